// VQVAE_gcn_47072841564919
// MI455X (gfx1250) — compile-verified
//
#include <hip/hip_runtime.h>
#include <stdint.h>

#define N_CODES   32768
#define EMB_DIM   256
#define NUM_EMB   4096
#define BETA      0.25f

#define WAVES_PER_BLOCK 8
#define ROWS_PER_WAVE   16
#define ROWS_PER_BLOCK  (WAVES_PER_BLOCK * ROWS_PER_WAVE)   // 128
#define BPITCH          260   // LDS row pitch (floats): 260%64==4 -> conflict-free WMMA-B reads
#define NCHUNK          (NUM_EMB / 16)                      // 256

typedef float v2f __attribute__((ext_vector_type(2)));
typedef float v4f __attribute__((ext_vector_type(4)));
typedef float v8f __attribute__((ext_vector_type(8)));

// ---------------------------------------------------------------------------
// Kernel 1: per-codebook-row squared norms ||w_k||^2  (4096 rows, tiny)
// ---------------------------------------------------------------------------
__global__ void vq_wnorm_kernel(const float* __restrict__ w,
                                float* __restrict__ wnorm) {
  int g = blockIdx.x * blockDim.x + threadIdx.x;      // 0..4095, one row/thread
  const v4f* row = (const v4f*)(w + (size_t)g * EMB_DIM);
  float s = 0.f;
#pragma unroll
  for (int i = 0; i < EMB_DIM / 4; ++i) {
    v4f v = row[i];
    s = fmaf(v.x, v.x, s);
    s = fmaf(v.y, v.y, s);
    s = fmaf(v.z, v.z, s);
    s = fmaf(v.w, v.w, s);
  }
  wnorm[g] = s;
}

// ---------------------------------------------------------------------------
// Kernel 2: fused distance-GEMM (fp32 WMMA) + argmin + gather + loss partials
//   - wave owns 16 rows of x; A-matrix lives in registers in WMMA layout
//   - codebook tiles double-buffered in LDS, filled by async global->LDS DMA
//     (ASYNCcnt path), one barrier per 16-column chunk
// ---------------------------------------------------------------------------
__global__ void __launch_bounds__(256)
vq_main_kernel(const float* __restrict__ x, const float* __restrict__ w,
               const float* __restrict__ wnorm,
               float* __restrict__ outq, float* __restrict__ outidx,
               float* __restrict__ partials) {
  __shared__ __align__(16) float bsm[2][16 * BPITCH];
  __shared__ int rowsel[WAVES_PER_BLOCK][ROWS_PER_WAVE];

  const int tid  = threadIdx.x;
  const int wid  = tid >> 5;
  const int lane = tid & 31;
  const int m    = lane & 15;   // M (A) / N (B) index within tile
  const int h    = lane >> 4;   // half selector (K pairing per WMMA f32 layout)
  const int row_base = blockIdx.x * ROWS_PER_BLOCK + wid * ROWS_PER_WAVE;

  // ---- per-thread async staging descriptors (4 x b128 per chunk) ----------
  // tile = 16 rows x 256 floats; float4 id q -> row q>>6, float4-col q&63
  uint64_t gbase[4];     // global address of this thread's piece in chunk 0
  uint32_t loff[4];      // LDS byte offset of that piece within a buffer
#pragma unroll
  for (int i = 0; i < 4; ++i) {
    int q  = tid + i * 256;
    int r  = q >> 6;
    int c4 = q & 63;
    gbase[i] = (uint64_t)(uintptr_t)(w + (size_t)r * EMB_DIM + c4 * 4);
    loff[i]  = (uint32_t)((r * BPITCH + c4 * 4) * sizeof(float));
  }
  const uint32_t lbuf0 = (uint32_t)(uintptr_t)(&bsm[0][0]);
  const uint32_t lbuf1 = (uint32_t)(uintptr_t)(&bsm[1][0]);

  // ---- load this wave's 16x256 A block into WMMA-A layout registers --------
  // lane(m,h), reg kb holds (A[m][4kb+2h], A[m][4kb+2h+1])
  v2f areg[64];
  {
    const float* xr = x + (size_t)(row_base + m) * EMB_DIM + 2 * h;
#pragma unroll
    for (int kb = 0; kb < 64; ++kb)
      areg[kb] = *(const v2f*)(xr + 4 * kb);
  }

  float best_s[8];
  int   best_j[8];
#pragma unroll
  for (int r = 0; r < 8; ++r) { best_s[r] = 3.4e38f; best_j[r] = 0; }

  // ---- prologue: DMA chunk 0 into buffer 0 --------------------------------
#pragma unroll
  for (int i = 0; i < 4; ++i) {
    uint64_t ga = gbase[i];
    uint32_t la = lbuf0 + loff[i];
    asm volatile("global_load_async_to_lds_b128 %0, %1, off"
                 :: "v"(la), "v"(ga) : "memory");
  }
  asm volatile("s_wait_asynccnt 0x0" ::: "memory");
  __syncthreads();

  // ---- sweep the codebook: compute chunk jc, DMA chunk jc+1 ---------------
  for (int jc = 0; jc < NCHUNK; ++jc) {
    const int j0  = jc * 16;
    const int buf = jc & 1;
    const int jn  = (jc + 1) & (NCHUNK - 1);     // wrap keeps last prefetch in-bounds
    const uint32_t lnext = (buf ? lbuf0 : lbuf1);
    const uint64_t goff  = (uint64_t)jn * (16u * EMB_DIM * sizeof(float));

    // issue async DMA for the next tile (overlaps with the WMMAs below)
#pragma unroll
    for (int i = 0; i < 4; ++i) {
      uint64_t ga = gbase[i] + goff;
      uint32_t la = lnext + loff[i];
      asm volatile("global_load_async_to_lds_b128 %0, %1, off"
                   :: "v"(la), "v"(ga) : "memory");
    }

    // 64 x V_WMMA_F32_16X16X4_F32 over D=256 -> dot(x_i, w_j) tile
    v8f c = {0.f, 0.f, 0.f, 0.f, 0.f, 0.f, 0.f, 0.f};
    const float* bb = &bsm[buf][m * BPITCH + 2 * h];
#pragma unroll
    for (int kb = 0; kb < 64; ++kb) {
      v2f b = *(const v2f*)(bb + 4 * kb);
      c = __builtin_amdgcn_wmma_f32_16x16x4_f32(
          /*neg_a=*/false, areg[kb], /*neg_b=*/false, b,
          /*c_mod=*/(short)0, c, /*reuse_a=*/false, /*reuse_b=*/false);
    }

    // score = ||w_j||^2 - 2*dot  (||x_i||^2 constant per row -> irrelevant)
    const float wn = wnorm[j0 + m];
    const int   jj = j0 + m;
#pragma unroll
    for (int r = 0; r < 8; ++r) {
      float s = fmaf(-2.f, c[r], wn);
      if (s < best_s[r]) { best_s[r] = s; best_j[r] = jj; }   // first-min wins ties
    }

    // this wave's DMA must land before anyone reads the next buffer
    asm volatile("s_wait_asynccnt 0x0" ::: "memory");
    __syncthreads();
  }

  // ---- argmin reduction across the 16 lanes of each half ------------------
#pragma unroll
  for (int r = 0; r < 8; ++r) {
    float s  = best_s[r];
    int   bj = best_j[r];
#pragma unroll
    for (int mask = 1; mask < 16; mask <<= 1) {
      float so = __shfl_xor(s, mask, 32);
      int   jo = __shfl_xor(bj, mask, 32);
      if (so < s || (so == s && jo < bj)) { s = so; bj = jo; }
    }
    if (m == 0) rowsel[wid][r + 8 * h] = bj;   // lanes 0 & 16 cover rows r, r+8
  }
  __syncthreads();

  // ---- gather winning rows, write quantized_st, accumulate (q-x)^2 --------
  float lsum = 0.f;
  for (int rr = 0; rr < ROWS_PER_WAVE; ++rr) {
    const int row = row_base + rr;
    const int bj  = rowsel[wid][rr];
    if (lane == 0) outidx[row] = (float)bj;
    const v4f* wr = (const v4f*)(w + (size_t)bj * EMB_DIM);
    const v4f* xr = (const v4f*)(x + (size_t)row * EMB_DIM);
    v4f*       qr = (v4f*)(outq + (size_t)row * EMB_DIM);
#pragma unroll
    for (int t = 0; t < EMB_DIM / (4 * 32); ++t) {
      const int ci = lane + t * 32;          // float4 index within row (0..63)
      v4f qv = wr[ci];
      v4f xv = xr[ci];
      qr[ci] = qv;                           // quantized_st == quantized (value-wise)
      v4f d = qv - xv;
      lsum = fmaf(d.x, d.x, lsum);
      lsum = fmaf(d.y, d.y, lsum);
      lsum = fmaf(d.z, d.z, lsum);
      lsum = fmaf(d.w, d.w, lsum);
    }
  }
#pragma unroll
  for (int mask = 1; mask < 32; mask <<= 1) lsum += __shfl_xor(lsum, mask, 32);
  if (lane == 0) partials[blockIdx.x * WAVES_PER_BLOCK + wid] = lsum;
}

// ---------------------------------------------------------------------------
// Kernel 3: reduce 2048 per-wave partials -> commitloss, contrastloss
// ---------------------------------------------------------------------------
__global__ void vq_loss_kernel(const float* __restrict__ partials,
                               float* __restrict__ outsc) {
  __shared__ float sm[256];
  float s = 0.f;
  for (int i = threadIdx.x; i < (N_CODES / ROWS_PER_BLOCK) * WAVES_PER_BLOCK; i += 256)
    s += partials[i];
  sm[threadIdx.x] = s;
  __syncthreads();
  for (int off = 128; off > 0; off >>= 1) {
    if (threadIdx.x < off) sm[threadIdx.x] += sm[threadIdx.x + off];
    __syncthreads();
  }
  if (threadIdx.x == 0) {
    float mse = sm[0] / (float)((size_t)N_CODES * EMB_DIM);
    outsc[0] = (1.f + BETA) * mse;   // both MSE terms are equal numerically
    outsc[1] = 0.f;                  // contrastloss
  }
}

// ---------------------------------------------------------------------------
extern "C" void kernel_launch(void* const* d_in, const int* in_sizes, int n_in,
                              void* d_out, int out_size, void* d_ws, size_t ws_size,
                              hipStream_t stream) {
  (void)in_sizes; (void)n_in; (void)out_size; (void)ws_size;
  const float* x = (const float*)d_in[0];   // [32768, 256]
  const float* w = (const float*)d_in[1];   // [4096, 256]

  float* out     = (float*)d_out;
  float* outq    = out;                                   // [N, D]
  float* outidx  = out + (size_t)N_CODES * EMB_DIM;       // [N]   (as float)
  float* outsc   = outidx + N_CODES;                      // commit, contrast

  float* wnorm    = (float*)d_ws;                         // [4096]
  float* partials = wnorm + NUM_EMB;                      // [2048]

  vq_wnorm_kernel<<<NUM_EMB / 256, 256, 0, stream>>>(w, wnorm);
  vq_main_kernel<<<N_CODES / ROWS_PER_BLOCK, 256, 0, stream>>>(
      x, w, wnorm, outq, outidx, partials);
  vq_loss_kernel<<<1, 256, 0, stream>>>(partials, outsc);
}